// _FFTConv_63548336111890
// MI455X (gfx1250) — compile-verified
//
#include <hip/hip_runtime.h>
#include <math.h>

// Problem constants (match reference)
#define B_   8
#define CIN  32
#define COUT 32
#define KK   3
#define H_   256
#define W_   256
#define HW   (H_*W_)

#define TWO_PI 6.283185307179586f

typedef __attribute__((ext_vector_type(2))) float v2f;
typedef __attribute__((ext_vector_type(8))) float v8f;

// gfx1250 async global->LDS path (guarded; host pass / older toolchains fall back)
#if defined(__has_builtin)
#  if __has_builtin(__builtin_amdgcn_global_load_async_to_lds_b32) && \
      __has_builtin(__builtin_amdgcn_s_wait_asynccnt)
#    define USE_ASYNC_LDS 1
#  endif
#endif
#ifndef USE_ASYNC_LDS
#  define USE_ASYNC_LDS 0
#endif

#if USE_ASYNC_LDS
typedef __attribute__((address_space(1))) int g_as_int;   // global (cuda_device)
typedef __attribute__((address_space(3))) int l_as_int;   // LDS (cuda_shared)
#endif

// ---------------------------------------------------------------------------
// Kernel 1: g[o][c][p][v] = sum_q conj(w)[o,c,p,q] * e^{+i 2pi v q / W}
// (separable half of conj(FFT2(padded 3x3 kernel))). 6.3 MB -> stays in L2.
// ---------------------------------------------------------------------------
__global__ void k_compute_g(const float* __restrict__ wr,
                            const float* __restrict__ wi,
                            float* __restrict__ g) {
    int idx = blockIdx.x * blockDim.x + threadIdx.x;   // idx = (oc*3 + p)*256 + v
    const int N = COUT * CIN * KK * W_;
    if (idx >= N) return;
    int v  = idx & (W_ - 1);
    int p  = (idx >> 8) % KK;
    int oc = idx / (KK * W_);
    float accr = 0.f, acci = 0.f;
#pragma unroll
    for (int q = 0; q < KK; ++q) {
        float ang = TWO_PI * (float)(v * q) / (float)W_;
        float s, c;
        __sincosf(ang, &s, &c);
        float a =  wr[oc * (KK * KK) + p * KK + q];
        float b = -wi[oc * (KK * KK) + p * KK + q];   // conjugate of weight
        accr += a * c - b * s;
        acci += a * s + b * c;
    }
    g[2 * idx]     = accr;
    g[2 * idx + 1] = acci;
}

// ---------------------------------------------------------------------------
// Kernel 2: materialize kernel_fr output (required return value, 536 MB write).
// kfr[o][c][u][v] = sum_p g[o][c][p][v] * e^{+i 2pi u p / H}
// ---------------------------------------------------------------------------
__global__ void k_write_kfr(const float* __restrict__ g,
                            float* __restrict__ kfr) {
    int idx = blockIdx.x * blockDim.x + threadIdx.x;   // [oc][u][v], exact grid
    int v  = idx & (W_ - 1);
    int u  = (idx >> 8) & (H_ - 1);
    int oc = idx >> 16;
    float kr = 0.f, ki = 0.f;
#pragma unroll
    for (int p = 0; p < KK; ++p) {
        float ang = TWO_PI * (float)(u * p) / (float)H_;
        float s, c;
        __sincosf(ang, &s, &c);
        int gb = ((oc * KK + p) << 9) + 2 * v;         // (oc*3+p)*512 + 2v
        float gr = g[gb], gi = g[gb + 1];
        kr += gr * c - gi * s;
        ki += gr * s + gi * c;
    }
    kfr[2 * idx]     = kr;
    kfr[2 * idx + 1] = ki;
}

// ---------------------------------------------------------------------------
// Kernel 3: per-pixel complex channel contraction via V_WMMA_F32_16X16X4_F32.
//   out_fr[b,o,u,v] = sum_c s[b,c,u,v] * kfr[o,c,u,v]
// Workgroup = 8 waves = 8 adjacent pixels of one row u.
//  - signal tile staged in LDS (async global->LDS where available)
//  - kfr tile computed once per workgroup from g into LDS
//  - one wave per pixel: M = batch (8 of 16 rows), N = o (2 tiles), K = c (8x4)
//  - complex product = 4 real WMMAs; batch mask is a multiply (EXEC stays full)
// ---------------------------------------------------------------------------
#define VBLK 8
#define SIG_CSTRIDE (VBLK * B_ + 8)        // 72 floats; pad kills bank conflicts
#define KFR_CSTRIDE (VBLK * COUT + 8)      // 264 floats; pad kills bank conflicts

__global__ void __launch_bounds__(256)
k_contract_wmma(const float* __restrict__ sr, const float* __restrict__ si,
                const float* __restrict__ g,  float* __restrict__ out) {
    __shared__ float s_sr[CIN * SIG_CSTRIDE];
    __shared__ float s_si[CIN * SIG_CSTRIDE];
    __shared__ float s_kr[CIN * KFR_CSTRIDE];
    __shared__ float s_ki[CIN * KFR_CSTRIDE];

    const int t    = threadIdx.x;
    const int lane = t & 31;
    const int wave = t >> 5;
    const int u  = blockIdx.x >> 5;            // 256 rows
    const int v0 = (blockIdx.x & 31) * VBLK;   // 32 v-blocks of 8

    // Row twiddles Tu[p] = e^{+i 2pi u p / H} (uniform across the block)
    float tur[KK], tui[KK];
#pragma unroll
    for (int p = 0; p < KK; ++p) {
        float ang = TWO_PI * (float)(u * p) / (float)H_;
        __sincosf(ang, &tui[p], &tur[p]);
    }

    // ---- stage signal tile: s[b, c, v0..v0+7] -> LDS [c][vo][b] ----
#pragma unroll
    for (int k = 0; k < 8; ++k) {              // 2048 elems / 256 threads
        int e  = t + k * 256;
        int vo = e & 7, c = (e >> 3) & 31, b = e >> 8;
        size_t gidx = ((size_t)(b * CIN + c) * H_ + u) * W_ + (v0 + vo);
        int    l    = c * SIG_CSTRIDE + vo * B_ + b;
#if USE_ASYNC_LDS
        __builtin_amdgcn_global_load_async_to_lds_b32(
            (g_as_int*)(sr + gidx), (l_as_int*)&s_sr[l], 0, 0);
        __builtin_amdgcn_global_load_async_to_lds_b32(
            (g_as_int*)(si + gidx), (l_as_int*)&s_si[l], 0, 0);
#else
        s_sr[l] = sr[gidx];
        s_si[l] = si[gidx];
#endif
    }

    // ---- compute kfr tile once per WG: kfr[o,c,v] = sum_p g[o,c,p,v]*Tu[p] ----
#pragma unroll
    for (int k = 0; k < 32; ++k) {             // 8192 elems / 256 threads
        int e  = t + k * 256;
        int vo = e & 7, c = (e >> 3) & 31, o = e >> 8;
        const float* gp = g + (size_t)((o * CIN + c) * KK) * (2 * W_) + 2 * (v0 + vo);
        float kr = 0.f, ki = 0.f;
#pragma unroll
        for (int p = 0; p < KK; ++p) {
            float gr = gp[p * 2 * W_];
            float gi = gp[p * 2 * W_ + 1];
            kr += gr * tur[p] - gi * tui[p];
            ki += gr * tui[p] + gi * tur[p];
        }
        int l = c * KFR_CSTRIDE + vo * COUT + o;
        s_kr[l] = kr;
        s_ki[l] = ki;
    }

#if USE_ASYNC_LDS
    __builtin_amdgcn_s_wait_asynccnt(0);
#endif
    __syncthreads();

    // ---- per-wave WMMA GEMM for pixel (u, v0+wave) ----
    const int   vo  = wave;
    const int   m   = lane & 15;               // matrix row (= batch when < 8)
    const int   khi = (lane >> 4) << 1;        // lanes 16..31 carry K rows +2
    const float msk = (m < B_) ? 1.0f : 0.0f;  // arithmetic mask, no EXEC games

    v8f accR[2] = {v8f{}, v8f{}};
    v8f accI[2] = {v8f{}, v8f{}};

#pragma unroll
    for (int ks = 0; ks < 8; ++ks) {
        const int c0 = ks * 4 + khi;

        // A operand from LDS signal tile
        const int sb = c0 * SIG_CSTRIDE + vo * B_ + (m & 7);
        float ar0 = s_sr[sb]               * msk;
        float ar1 = s_sr[sb + SIG_CSTRIDE] * msk;   // channel c0+1
        float ai0 = s_si[sb]               * msk;
        float ai1 = s_si[sb + SIG_CSTRIDE] * msk;
        v2f Ar  = { ar0,  ar1};
        v2f Ai  = { ai0,  ai1};
        v2f AiN = {-ai0, -ai1};

#pragma unroll
        for (int nt = 0; nt < 2; ++nt) {
            const int o = nt * 16 + m;
            const int kb = c0 * KFR_CSTRIDE + vo * COUT + o;
            v2f Br = { s_kr[kb], s_kr[kb + KFR_CSTRIDE] };
            v2f Bi = { s_ki[kb], s_ki[kb + KFR_CSTRIDE] };
            // (Ar + i Ai) * (Br + i Bi):
            accR[nt] = __builtin_amdgcn_wmma_f32_16x16x4_f32(false, Ar,  false, Br, (short)0, accR[nt], false, false);
            accR[nt] = __builtin_amdgcn_wmma_f32_16x16x4_f32(false, AiN, false, Bi, (short)0, accR[nt], false, false);
            accI[nt] = __builtin_amdgcn_wmma_f32_16x16x4_f32(false, Ar,  false, Bi, (short)0, accI[nt], false, false);
            accI[nt] = __builtin_amdgcn_wmma_f32_16x16x4_f32(false, Ai,  false, Br, (short)0, accI[nt], false, false);
        }
    }

    // ---- epilogue: VGPR r holds M=r for lanes 0..15 (N = lane) ----
    const int v = v0 + vo;
    if (lane < 16) {
#pragma unroll
        for (int r = 0; r < B_; ++r) {         // r = batch b (rows 8..15 junk)
#pragma unroll
            for (int nt = 0; nt < 2; ++nt) {
                int o = nt * 16 + lane;
                size_t oidx = (((size_t)(r * COUT + o) * H_ + u) * W_ + v) * 2;
                float2 val;
                val.x = accR[nt][r];
                val.y = accI[nt][r];
                *(float2*)(out + oidx) = val;  // 8-byte store (re, im)
            }
        }
    }
}

// ---------------------------------------------------------------------------
// Kernels 4/5: in-place 256-point inverse FFT (radix-2, in LDS).
// 128 threads = 128 butterflies/stage, 8 stages, e^{+2pi i} twiddles.
// Row pass applies the full 1/65536 scale; column pass is strided.
// ---------------------------------------------------------------------------
__device__ __forceinline__ void ifft256_core(float* re, float* im, int t) {
#pragma unroll
    for (int s = 1; s <= 8; ++s) {
        const int mlen = 1 << s, half = mlen >> 1;
        const int j  = t & (half - 1);
        const int i1 = ((t >> (s - 1)) << s) + j;
        const int i2 = i1 + half;
        float ang = TWO_PI * (float)j / (float)mlen;
        float sn, cs;
        __sincosf(ang, &sn, &cs);
        float xr = re[i2], xi = im[i2];
        float tr = xr * cs - xi * sn;
        float ti = xr * sn + xi * cs;
        float ur = re[i1], ui = im[i1];
        re[i1] = ur + tr;  im[i1] = ui + ti;
        re[i2] = ur - tr;  im[i2] = ui - ti;
        __syncthreads();
    }
}

__global__ void __launch_bounds__(128)
k_ifft_rows(float* __restrict__ data) {
    __shared__ float re[256];
    __shared__ float im[256];
    float* base = data + (size_t)blockIdx.x * (2 * W_);   // one row (b,o,u)
    const int t = threadIdx.x;
#pragma unroll
    for (int e = 0; e < 2; ++e) {
        int i  = t + e * 128;
        int br = __brev((unsigned)i) >> 24;               // bit-reverse load
        re[br] = base[2 * i];
        im[br] = base[2 * i + 1];
    }
    __syncthreads();
    ifft256_core(re, im, t);
    const float sc = 1.0f / 65536.0f;                     // full 1/N^2 here
#pragma unroll
    for (int e = 0; e < 2; ++e) {
        int i = t + e * 128;
        base[2 * i]     = re[i] * sc;
        base[2 * i + 1] = im[i] * sc;
    }
}

__global__ void __launch_bounds__(128)
k_ifft_cols(float* __restrict__ data) {
    __shared__ float re[256];
    __shared__ float im[256];
    const int img = blockIdx.x >> 8;                      // (b*32 + o)
    const int v   = blockIdx.x & (W_ - 1);
    float* base = data + (size_t)img * (2 * HW) + 2 * v;  // column v, stride 2*W_
    const int t = threadIdx.x;
#pragma unroll
    for (int e = 0; e < 2; ++e) {
        int i  = t + e * 128;
        int br = __brev((unsigned)i) >> 24;
        re[br] = base[(size_t)i * (2 * W_)];
        im[br] = base[(size_t)i * (2 * W_) + 1];
    }
    __syncthreads();
    ifft256_core(re, im, t);
#pragma unroll
    for (int e = 0; e < 2; ++e) {
        int i = t + e * 128;
        base[(size_t)i * (2 * W_)]     = re[i];
        base[(size_t)i * (2 * W_) + 1] = im[i];
    }
}

// ---------------------------------------------------------------------------
extern "C" void kernel_launch(void* const* d_in, const int* in_sizes, int n_in,
                              void* d_out, int out_size, void* d_ws, size_t ws_size,
                              hipStream_t stream) {
    const float* sr = (const float*)d_in[0];   // signal_real  [8,32,256,256]
    const float* si = (const float*)d_in[1];   // signal_imag  [8,32,256,256]
    const float* wr = (const float*)d_in[2];   // weight_real  [32,32,3,3]
    const float* wi = (const float*)d_in[3];   // weight_imag  [32,32,3,3]

    float* out = (float*)d_out;                                // output, complex interleaved
    float* kfr = out + (size_t)2 * B_ * COUT * HW;             // kernel_fr, complex interleaved
    float* g   = (float*)d_ws;                                 // 32*32*3*256 complex (6.3 MB)

    // 1) separable half-transform of conj(weight)
    k_compute_g<<<(COUT * CIN * KK * W_ + 255) / 256, 256, 0, stream>>>(wr, wi, g);
    // 2) write kernel_fr output (536 MB, the dominant traffic)
    k_write_kfr<<<(COUT * CIN * HW) / 256, 256, 0, stream>>>(g, kfr);
    // 3) WMMA channel contraction -> output_fr written in-place into d_out
    k_contract_wmma<<<HW / VBLK, 256, 0, stream>>>(sr, si, g, out);
    // 4) inverse FFT rows (in-place, scaled), then columns (in-place)
    k_ifft_rows<<<B_ * COUT * H_, 128, 0, stream>>>(out);
    k_ifft_cols<<<B_ * COUT * W_, 128, 0, stream>>>(out);
}